// GATTemporalNet_2078764172107
// MI455X (gfx1250) — compile-verified
//
#include <hip/hip_runtime.h>
#include <hip/hip_bf16.h>
#include <math.h>

// ---------------- problem constants (match reference) ----------------
#define NN     32768
#define EE     262144
#define BBATCH 16
#define MM     2048
#define DIN    16
#define HH     4
#define CC     64
#define FF     256
#define ETOT   (EE + NN)   // edges + self loops

typedef float v2f __attribute__((ext_vector_type(2)));
typedef float v8f __attribute__((ext_vector_type(8)));

// CDNA5 fp32 WMMA: D(16x16,f32) = A(16x4,f32) x B(4x16,f32) + C
__device__ __forceinline__ v8f wmma4_f32(v2f a, v2f b, v8f c) {
  return __builtin_amdgcn_wmma_f32_16x16x4_f32(false, a, false, b, (short)0, c,
                                               false, false);
}

// ---------------- utility kernels ----------------
__global__ void fill_kernel(float* __restrict__ p, float v, int n) {
  int i = blockIdx.x * blockDim.x + threadIdx.x;
  if (i < n) p[i] = v;
}

__device__ __forceinline__ void atomicMaxFloat(float* addr, float val) {
  unsigned int* ua = (unsigned int*)addr;
  unsigned int old = __hip_atomic_load(ua, __ATOMIC_RELAXED, __HIP_MEMORY_SCOPE_AGENT);
  while (__uint_as_float(old) < val) {
    unsigned int assumed = old;
    old = atomicCAS(ua, assumed, __float_as_uint(val));
    if (old == assumed) break;
  }
}

// ---------------- generic WMMA f32 GEMM: Out[M,N] = act(A[M,K] @ W[K,N] + bias) ----------------
// one wave32 per 16x16 output tile; K stepped by 4 through V_WMMA_F32_16X16X4_F32
__global__ void wmma_gemm_f32(const float* __restrict__ A, const float* __restrict__ W,
                              const float* __restrict__ bias, float* __restrict__ Out,
                              int Mdim, int K, int Ndim, int act) {
  int wave = (blockIdx.x * blockDim.x + threadIdx.x) >> 5;
  int lane = threadIdx.x & 31;
  int ntn = Ndim >> 4;
  int ntm = Mdim >> 4;
  if (wave >= ntm * ntn) return;           // wave-uniform: EXEC stays all-ones
  int tm = wave / ntn, tn = wave % ntn;
  int m  = tm * 16 + (lane & 15);          // A: lanes 0-15 rows, both halves same M
  int n  = tn * 16 + (lane & 15);          // B/C: N striped across lane half
  int kb = (lane >> 4) * 2;                // half-wave selects K pair {0,1} vs {2,3}
  v8f c = {};
  for (int k0 = 0; k0 < K; k0 += 4) {
    v2f a, b;
    a.x = A[(size_t)m * K + k0 + kb];
    a.y = A[(size_t)m * K + k0 + kb + 1];
    b.x = W[(size_t)(k0 + kb) * Ndim + n];
    b.y = W[(size_t)(k0 + kb + 1) * Ndim + n];
    c = wmma4_f32(a, b, c);
  }
  int r0 = tm * 16 + (lane >> 4) * 8;      // C/D: VGPR r -> row r (+8 for high lanes)
  float bv = bias ? bias[n] : 0.0f;
  for (int r = 0; r < 8; ++r) {
    float y = c[r] + bv;
    if (act == 1) y = fmaxf(y, 0.0f);
    Out[(size_t)(r0 + r) * Ndim + n] = y;
  }
}

// ---------------- edge statistics: deg & mean incoming edge_attr (self-loop fill) ----------------
__global__ void edge_stats_kernel(const int* __restrict__ ei, const float* __restrict__ eattr,
                                  float* __restrict__ deg, float* __restrict__ sattr) {
  int e = blockIdx.x * blockDim.x + threadIdx.x;
  if (e >= EE) return;
  int t = ei[EE + e];
  atomicAdd(&deg[t], 1.0f);
  atomicAdd(&sattr[t * 3 + 0], eattr[e * 3 + 0]);
  atomicAdd(&sattr[t * 3 + 1], eattr[e * 3 + 1]);
  atomicAdd(&sattr[t * 3 + 2], eattr[e * 3 + 2]);
}

__global__ void sattr_div_kernel(float* __restrict__ sattr, const float* __restrict__ deg) {
  int n = blockIdx.x * blockDim.x + threadIdx.x;
  if (n >= NN) return;
  float inv = 1.0f / fmaxf(deg[n], 1.0f);
  sattr[n * 3 + 0] *= inv;
  sattr[n * 3 + 1] *= inv;
  sattr[n * 3 + 2] *= inv;
}

// ---------------- GATv2 logits: leaky_relu(xl[s]+xr[t]+ea) . att  (per head) ----------------
__global__ void gat_logits_kernel(const float* __restrict__ xl, const float* __restrict__ xr,
                                  const int* __restrict__ ei, const float* __restrict__ eattr,
                                  const float* __restrict__ sattr, const float* __restrict__ we,
                                  const float* __restrict__ att, float* __restrict__ logits,
                                  float* __restrict__ lmax) {
  int e = blockIdx.x;
  int c = threadIdx.x;            // channel 0..255, head = c>>6
  int s, t; const float* ea;
  if (e < EE) { s = ei[e]; t = ei[EE + e]; ea = eattr + (size_t)e * 3; }
  else        { s = e - EE; t = s;         ea = sattr + (size_t)s * 3; }
  float eav = ea[0] * we[c] + ea[1] * we[FF + c] + ea[2] * we[2 * FF + c];
  float mv  = xl[(size_t)s * FF + c] + xr[(size_t)t * FF + c] + eav;
  mv = (mv > 0.0f) ? mv : 0.2f * mv;            // LeakyReLU(0.2)
  float p = mv * att[c];                        // att is [H,C] flat == 256
  // wave32 tree reduce, then cross-wave combine in LDS (8 waves/block)
  for (int off = 16; off > 0; off >>= 1) p += __shfl_down(p, off, 32);
  __shared__ float red[8];
  if ((c & 31) == 0) red[c >> 5] = p;
  __syncthreads();
  if (c < HH) {
    float lg = red[2 * c] + red[2 * c + 1];     // head h = waves 2h,2h+1
    logits[(size_t)e * HH + c] = lg;
    atomicMaxFloat(&lmax[(size_t)t * HH + c], lg);
  }
}

// exp(logit - lmax[t]) in place + segment sum
__global__ void gat_exp_kernel(float* __restrict__ logits, const float* __restrict__ lmax,
                               const int* __restrict__ ei, float* __restrict__ sumex) {
  int idx = blockIdx.x * blockDim.x + threadIdx.x;
  if (idx >= ETOT * HH) return;
  int e = idx >> 2, h = idx & 3;
  int t = (e < EE) ? ei[EE + e] : (e - EE);
  float ex = __expf(logits[idx] - lmax[(size_t)t * HH + h]);
  logits[idx] = ex;
  atomicAdd(&sumex[(size_t)t * HH + h], ex);
}

// out[t] += alpha * xl[s]
__global__ void gat_scatter_kernel(const float* __restrict__ exv, const float* __restrict__ sumex,
                                   const float* __restrict__ xl, const int* __restrict__ ei,
                                   float* __restrict__ out) {
  int e = blockIdx.x;
  int c = threadIdx.x;
  int s, t;
  if (e < EE) { s = ei[e]; t = ei[EE + e]; }
  else        { s = e - EE; t = s; }
  int h = c >> 6;
  float alpha = exv[(size_t)e * HH + h] / sumex[(size_t)t * HH + h];
  atomicAdd(&out[(size_t)t * FF + c], alpha * xl[(size_t)s * FF + c]);
}

// ---------------- bias + LayerNorm + ELU (+ optional residual) ----------------
__global__ void ln_elu_kernel(const float* __restrict__ attout, const float* __restrict__ gbias,
                              const float* __restrict__ g, const float* __restrict__ bvec,
                              const float* __restrict__ residual, float* __restrict__ out) {
  int n = blockIdx.x, c = threadIdx.x;
  float y = attout[(size_t)n * FF + c] + gbias[c];
  __shared__ float red[8];
  __shared__ float stat[2];
  float s = y;
  for (int off = 16; off > 0; off >>= 1) s += __shfl_down(s, off, 32);
  if ((c & 31) == 0) red[c >> 5] = s;
  __syncthreads();
  if (c == 0) {
    float m = 0.f; for (int i = 0; i < 8; ++i) m += red[i];
    stat[0] = m * (1.0f / FF);
  }
  __syncthreads();
  float mu = stat[0];
  float d  = y - mu;
  float sq = d * d;
  for (int off = 16; off > 0; off >>= 1) sq += __shfl_down(sq, off, 32);
  if ((c & 31) == 0) red[c >> 5] = sq;
  __syncthreads();
  if (c == 0) {
    float v = 0.f; for (int i = 0; i < 8; ++i) v += red[i];
    stat[1] = v * (1.0f / FF);
  }
  __syncthreads();
  float v = d * rsqrtf(stat[1] + 1e-5f) * g[c] + bvec[c];
  v = (v > 0.0f) ? v : expm1f(v);               // ELU
  if (residual) v += residual[(size_t)n * FF + c];
  out[(size_t)n * FF + c] = v;
}

// ---------------- conv1d(K=3,'SAME') as 3 shifted WMMA GEMMs, fused BN/ReLU/residual ----------------
// X node-major [NN,FF]; Wc [O=FF, I=FF, K=3]; Out node-major [NN,FF]
__global__ void tcn_conv_wmma(const float* __restrict__ X, const float* __restrict__ Wc,
                              const float* __restrict__ bias, const float* __restrict__ bng,
                              const float* __restrict__ bnb, const float* __restrict__ bnm,
                              const float* __restrict__ bnv, const float* __restrict__ residual,
                              float* __restrict__ Out) {
  int wave = (blockIdx.x * blockDim.x + threadIdx.x) >> 5;
  int lane = threadIdx.x & 31;
  const int ntn = FF >> 4;                       // 16
  if (wave >= (NN >> 4) * ntn) return;
  int tm = wave / ntn, tn = wave % ntn;
  int mrow = tm * 16 + (lane & 15);
  int n    = tn * 16 + (lane & 15);
  int kb   = (lane >> 4) * 2;
  int mloc = mrow & (MM - 1);                    // position within graph (tiles never straddle)
  v8f c = {};
  for (int k = 0; k < 3; ++k) {
    int dk = k - 1;
    int ok = (mloc + dk >= 0) && (mloc + dk < MM);
    float mask = ok ? 1.0f : 0.0f;
    const float* xrow = X + (size_t)(ok ? (mrow + dk) : mrow) * FF;   // never OOB
    const float* wk   = Wc + (size_t)n * (FF * 3) + k;                // Wk^T[i][n]
    for (int k0 = 0; k0 < FF; k0 += 4) {
      v2f a, b;
      a.x = xrow[k0 + kb] * mask;
      a.y = xrow[k0 + kb + 1] * mask;
      b.x = wk[(size_t)(k0 + kb) * 3];
      b.y = wk[(size_t)(k0 + kb + 1) * 3];
      c = wmma4_f32(a, b, c);
    }
  }
  int r0 = tm * 16 + (lane >> 4) * 8;
  float bv    = bias[n];
  float scale = rsqrtf(bnv[n] + 1e-5f) * bng[n];
  float mean  = bnm[n];
  float shift = bnb[n];
  for (int r = 0; r < 8; ++r) {
    int row = r0 + r;
    float y = (c[r] + bv - mean) * scale + shift;
    if (residual) y += residual[(size_t)row * FF + n];
    y = fmaxf(y, 0.0f);
    Out[(size_t)row * FF + n] = y;
  }
}

// ---------------- per-graph mean/max pooling over M: gemb[b, 0:F]=mean, [F:2F]=max ----------------
__global__ void pool_kernel(const float* __restrict__ tout, float* __restrict__ gemb) {
  int b = blockIdx.x, f = threadIdx.x;
  const float* base = tout + (size_t)b * MM * FF + f;
  float sum = 0.0f, mx = -3.402823466e38f;
  for (int m = 0; m < MM; ++m) {
    float v = base[(size_t)m * FF];               // coalesced: 256 lanes, consecutive f
    sum += v;
    mx = fmaxf(mx, v);
  }
  gemb[(size_t)b * (2 * FF) + f]      = sum * (1.0f / MM);
  gemb[(size_t)b * (2 * FF) + FF + f] = mx;
}

// ---------------- final head: out[b] = z2[b] . c3_w + c3_b ----------------
__global__ void head_final_kernel(const float* __restrict__ z2, const float* __restrict__ w,
                                  const float* __restrict__ b, float* __restrict__ out) {
  int t = threadIdx.x;
  if (t < BBATCH) {
    float acc = b[0];
    for (int i = 0; i < 64; ++i) acc += z2[t * 64 + i] * w[i];
    out[t] = acc;
  }
}

// ---------------- host orchestration ----------------
static inline int cdiv(int a, int b) { return (a + b - 1) / b; }

extern "C" void kernel_launch(void* const* d_in, const int* in_sizes, int n_in,
                              void* d_out, int out_size, void* d_ws, size_t ws_size,
                              hipStream_t stream) {
  const float* x       = (const float*)d_in[0];
  const int*   ei      = (const int*)  d_in[1];
  const float* eattr   = (const float*)d_in[2];
  const float* proj_w  = (const float*)d_in[4];
  const float* proj_b  = (const float*)d_in[5];
  const float* g1_wl   = (const float*)d_in[6];
  const float* g1_bl   = (const float*)d_in[7];
  const float* g1_wr   = (const float*)d_in[8];
  const float* g1_br   = (const float*)d_in[9];
  const float* g1_we   = (const float*)d_in[10];
  const float* g1_att  = (const float*)d_in[11];
  const float* g1_bias = (const float*)d_in[12];
  const float* ln1_g   = (const float*)d_in[13];
  const float* ln1_b   = (const float*)d_in[14];
  const float* g2_wl   = (const float*)d_in[15];
  const float* g2_bl   = (const float*)d_in[16];
  const float* g2_wr   = (const float*)d_in[17];
  const float* g2_br   = (const float*)d_in[18];
  const float* g2_we   = (const float*)d_in[19];
  const float* g2_att  = (const float*)d_in[20];
  const float* g2_bias = (const float*)d_in[21];
  const float* ln2_g   = (const float*)d_in[22];
  const float* ln2_b   = (const float*)d_in[23];
  const float* tcn_w1  = (const float*)d_in[24];
  const float* tcn_b1  = (const float*)d_in[25];
  const float* bn1_g   = (const float*)d_in[26];
  const float* bn1_b   = (const float*)d_in[27];
  const float* bn1_m   = (const float*)d_in[28];
  const float* bn1_v   = (const float*)d_in[29];
  const float* tcn_w2  = (const float*)d_in[30];
  const float* tcn_b2  = (const float*)d_in[31];
  const float* bn2_g   = (const float*)d_in[32];
  const float* bn2_b   = (const float*)d_in[33];
  const float* bn2_m   = (const float*)d_in[34];
  const float* bn2_v   = (const float*)d_in[35];
  const float* c1_w    = (const float*)d_in[36];
  const float* c1_b    = (const float*)d_in[37];
  const float* c2_w    = (const float*)d_in[38];
  const float* c2_b    = (const float*)d_in[39];
  const float* c3_w    = (const float*)d_in[40];
  const float* c3_b    = (const float*)d_in[41];
  float* out = (float*)d_out;

  // workspace carve-up (floats)
  float* ws = (float*)d_ws;
  size_t o = 0;
  float* h0     = ws + o; o += (size_t)NN * 64;
  float* xl     = ws + o; o += (size_t)NN * FF;
  float* xr     = ws + o; o += (size_t)NN * FF;
  float* attout = ws + o; o += (size_t)NN * FF;
  float* h1     = ws + o; o += (size_t)NN * FF;
  float* h2     = ws + o; o += (size_t)NN * FF;
  float* logits = ws + o; o += (size_t)ETOT * HH;
  float* lmax   = ws + o; o += (size_t)NN * HH;
  float* sumex  = ws + o; o += (size_t)NN * HH;
  float* deg    = ws + o; o += (size_t)NN;
  float* sattr  = ws + o; o += (size_t)NN * 3;
  float* gemb   = ws + o; o += (size_t)BBATCH * 2 * FF;
  float* z1     = ws + o; o += (size_t)BBATCH * FF;
  float* z2     = ws + o; o += (size_t)BBATCH * 64;
  float* t1   = xl;   // reuse after GAT layers
  float* tout = xr;

  const int TPB = 256;
  // --- self-loop edge_attr stats (shared by both layers) ---
  fill_kernel<<<cdiv(NN, TPB), TPB, 0, stream>>>(deg, 0.0f, NN);
  fill_kernel<<<cdiv(NN * 3, TPB), TPB, 0, stream>>>(sattr, 0.0f, NN * 3);
  edge_stats_kernel<<<cdiv(EE, TPB), TPB, 0, stream>>>(ei, eattr, deg, sattr);
  sattr_div_kernel<<<cdiv(NN, TPB), TPB, 0, stream>>>(sattr, deg);

  // --- input projection: h0 = x @ proj_w + proj_b   [N,16]@[16,64] ---
  {
    int tiles = (NN / 16) * (64 / 16);
    wmma_gemm_f32<<<cdiv(tiles, 8), TPB, 0, stream>>>(x, proj_w, proj_b, h0, NN, DIN, 64, 0);
  }

  // --- GAT layer 1 (64 -> 256) ---
  {
    int tiles = (NN / 16) * (FF / 16);
    wmma_gemm_f32<<<cdiv(tiles, 8), TPB, 0, stream>>>(h0, g1_wl, g1_bl, xl, NN, 64, FF, 0);
    wmma_gemm_f32<<<cdiv(tiles, 8), TPB, 0, stream>>>(h0, g1_wr, g1_br, xr, NN, 64, FF, 0);
    fill_kernel<<<cdiv(NN * HH, TPB), TPB, 0, stream>>>(lmax, -3.402823466e38f, NN * HH);
    fill_kernel<<<cdiv(NN * HH, TPB), TPB, 0, stream>>>(sumex, 0.0f, NN * HH);
    fill_kernel<<<cdiv(NN * FF, TPB), TPB, 0, stream>>>(attout, 0.0f, NN * FF);
    gat_logits_kernel<<<ETOT, TPB, 0, stream>>>(xl, xr, ei, eattr, sattr, g1_we, g1_att, logits, lmax);
    gat_exp_kernel<<<cdiv(ETOT * HH, TPB), TPB, 0, stream>>>(logits, lmax, ei, sumex);
    gat_scatter_kernel<<<ETOT, TPB, 0, stream>>>(logits, sumex, xl, ei, attout);
    ln_elu_kernel<<<NN, FF, 0, stream>>>(attout, g1_bias, ln1_g, ln1_b, nullptr, h1);
  }

  // --- GAT layer 2 (256 -> 256, residual) ---
  {
    int tiles = (NN / 16) * (FF / 16);
    wmma_gemm_f32<<<cdiv(tiles, 8), TPB, 0, stream>>>(h1, g2_wl, g2_bl, xl, NN, FF, FF, 0);
    wmma_gemm_f32<<<cdiv(tiles, 8), TPB, 0, stream>>>(h1, g2_wr, g2_br, xr, NN, FF, FF, 0);
    fill_kernel<<<cdiv(NN * HH, TPB), TPB, 0, stream>>>(lmax, -3.402823466e38f, NN * HH);
    fill_kernel<<<cdiv(NN * HH, TPB), TPB, 0, stream>>>(sumex, 0.0f, NN * HH);
    fill_kernel<<<cdiv(NN * FF, TPB), TPB, 0, stream>>>(attout, 0.0f, NN * FF);
    gat_logits_kernel<<<ETOT, TPB, 0, stream>>>(xl, xr, ei, eattr, sattr, g2_we, g2_att, logits, lmax);
    gat_exp_kernel<<<cdiv(ETOT * HH, TPB), TPB, 0, stream>>>(logits, lmax, ei, sumex);
    gat_scatter_kernel<<<ETOT, TPB, 0, stream>>>(logits, sumex, xl, ei, attout);
    ln_elu_kernel<<<NN, FF, 0, stream>>>(attout, g2_bias, ln2_g, ln2_b, h1, h2);
  }

  // --- temporal conv stack (node-major layouts; conv as 3 shifted WMMA GEMMs) ---
  {
    int tiles = (NN / 16) * (FF / 16);
    tcn_conv_wmma<<<cdiv(tiles, 8), TPB, 0, stream>>>(h2, tcn_w1, tcn_b1, bn1_g, bn1_b, bn1_m, bn1_v,
                                                      nullptr, t1);
    tcn_conv_wmma<<<cdiv(tiles, 8), TPB, 0, stream>>>(t1, tcn_w2, tcn_b2, bn2_g, bn2_b, bn2_m, bn2_v,
                                                      h2, tout);
  }

  // --- pooling + MLP head ---
  pool_kernel<<<BBATCH, FF, 0, stream>>>(tout, gemb);
  {
    int tiles = (BBATCH / 16) * (FF / 16);  // 16 tiles
    wmma_gemm_f32<<<cdiv(tiles, 8), TPB, 0, stream>>>(gemb, c1_w, c1_b, z1, BBATCH, 2 * FF, FF, 1);
  }
  {
    int tiles = (BBATCH / 16) * (64 / 16);  // 4 tiles
    wmma_gemm_f32<<<cdiv(tiles, 8), TPB, 0, stream>>>(z1, c2_w, c2_b, z2, BBATCH, FF, 64, 1);
  }
  head_final_kernel<<<1, 32, 0, stream>>>(z2, c3_w, c3_b, out);
}